// MahLoss_4509715660840
// MI455X (gfx1250) — compile-verified
//
#include <hip/hip_runtime.h>
#include <hip/hip_bf16.h>

// ---------------------------------------------------------------------------
// Mahalanobis loss:  mean_{b,n} 0.5 * (g-p)^T Sigma (g-p)
//   pred [B,N,2] f32, gt [B,N,2] f32, sigma [B,N,2,2] f32  ->  scalar f32
// Memory-bound streaming reduction (~268 MB read, roofline ~11.5us @23.3TB/s).
// Stage 1: b128 nontemporal streaming + in-register quadratic form + block
//          reduction -> one f32 partial per block in d_ws (deterministic).
// Stage 2: single wave32 reduces 2048 partials with chained
//          V_WMMA_F32_16X16X4_F32 (B = all-ones => accumulates row sums,
//          layout-invariant for B, exact f32 arithmetic).
// ---------------------------------------------------------------------------

typedef float f4  __attribute__((ext_vector_type(4)));
typedef float f2  __attribute__((ext_vector_type(2)));
typedef float v8f __attribute__((ext_vector_type(8)));

#define NBLOCKS  2048          // multiple of 64 -> exact 16x4 tiling in stage 2
#define NTHREADS 256           // 8 wave32 per block

__global__ __launch_bounds__(NTHREADS)
void mah_partial_kernel(const f4* __restrict__ pred4,
                        const f4* __restrict__ gt4,
                        const f4* __restrict__ sig4,
                        float* __restrict__ partials,
                        int npairs)            // npairs = B*N/2
{
    float acc = 0.0f;
    const int stride = gridDim.x * blockDim.x;
    for (int p = blockIdx.x * blockDim.x + threadIdx.x; p < npairs; p += stride) {
        // 2 landmarks per iteration: pred/gt float4 = (x0,y0,x1,y1),
        // sigma two float4 = (A00,A01,A10,A11) per landmark. All b128 loads.
        f4 pr = __builtin_nontemporal_load(&pred4[p]);
        f4 gt = __builtin_nontemporal_load(&gt4[p]);
        f4 m0 = __builtin_nontemporal_load(&sig4[2 * p + 0]);
        f4 m1 = __builtin_nontemporal_load(&sig4[2 * p + 1]);

        float sx0 = gt.x - pr.x, sy0 = gt.y - pr.y;
        float sx1 = gt.z - pr.z, sy1 = gt.w - pr.w;

        // s^T Sigma s = sx^2*A00 + sx*sy*(A01+A10) + sy^2*A11
        float q0 = fmaf(sx0 * sx0, m0.x,
                   fmaf(sx0 * sy0, m0.y + m0.z,
                        sy0 * sy0 * m0.w));
        float q1 = fmaf(sx1 * sx1, m1.x,
                   fmaf(sx1 * sy1, m1.y + m1.z,
                        sy1 * sy1 * m1.w));
        acc += q0 + q1;
    }

    // wave32 reduction
    for (int off = 16; off > 0; off >>= 1)
        acc += __shfl_xor(acc, off, 32);

    __shared__ float smem[NTHREADS / 32];
    const int lane = threadIdx.x & 31;
    const int wave = threadIdx.x >> 5;
    if (lane == 0) smem[wave] = acc;
    __syncthreads();

    if (wave == 0) {
        float v = (lane < (NTHREADS / 32)) ? smem[lane] : 0.0f;
        for (int off = 4; off > 0; off >>= 1)
            v += __shfl_xor(v, off, 32);
        if (lane == 0) partials[blockIdx.x] = v;   // every block writes its slot
    }
}

// Stage 2: one wave32. Sum NBLOCKS f32 partials via chained f32 WMMA.
// A-matrix (16x4 f32) VGPR layout per ISA: lanes 0-15 hold (M=lane, K=0..1)
// in VGPR0..1; lanes 16-31 hold (M=lane-16, K=2..3). So lane l supplies
// elements v[(l&15)*4 + (l>>4)*2 + {0,1}] of a 64-value tile -> one b64 load.
// B = all ones (layout-invariant). D[m][n] = sum_k A[m][k] + C[m][n].
__global__ __launch_bounds__(32)
void mah_final_kernel(const float* __restrict__ partials,
                      float* __restrict__ out,
                      float scale)
{
    const int lane = threadIdx.x;          // 0..31, EXEC all ones (no divergence)
    const f2* p2 = (const f2*)partials;

    v8f c = {};                            // f32 accumulator tile, starts at 0
    f2 ones; ones.x = 1.0f; ones.y = 1.0f;

    const int laneoff = (lane & 15) * 2 + (lane >> 4);   // in float2 units

    #pragma unroll
    for (int t = 0; t < NBLOCKS / 64; ++t) {
        f2 a = p2[t * 32 + laneoff];
        // D = A(16x4) * ones(4x16) + C  -> accumulates per-row sums of tile
        c = __builtin_amdgcn_wmma_f32_16x16x4_f32(
                /*neg_a=*/false, a,
                /*neg_b=*/false, ones,
                /*c_mod=*/(short)0, c,
                /*reuse_a=*/false, /*reuse_b=*/false);
    }

    // lane n (n<16): VGPR v holds rowsum(M=v); lane n+16: rowsum(M=v+8).
    float s = c[0] + c[1] + c[2] + c[3] + c[4] + c[5] + c[6] + c[7];
    s += __shfl_xor(s, 16, 32);            // rows 0..7 + rows 8..15 = total
    if (lane == 0) out[0] = s * scale;
}

extern "C" void kernel_launch(void* const* d_in, const int* in_sizes, int n_in,
                              void* d_out, int out_size, void* d_ws, size_t ws_size,
                              hipStream_t stream)
{
    const f4* pred4 = (const f4*)d_in[0];   // pred_landmarks [B,N,2] f32
    const f4* gt4   = (const f4*)d_in[1];   // ground_truth   [B,N,2] f32
    const f4* sig4  = (const f4*)d_in[2];   // sigma        [B,N,2,2] f32
    float* out      = (float*)d_out;
    float* partials = (float*)d_ws;         // NBLOCKS f32, always fully written

    const int n_bn   = in_sizes[0] / 2;     // B*N landmarks
    const int npairs = in_sizes[0] / 4;     // landmark pairs (B*N even)
    const float scale = 0.5f / (float)n_bn; // fold 0.5 and the double mean

    mah_partial_kernel<<<NBLOCKS, NTHREADS, 0, stream>>>(pred4, gt4, sig4,
                                                         partials, npairs);
    mah_final_kernel<<<1, 32, 0, stream>>>(partials, out, scale);
}